// DGI_6528350290006
// MI455X (gfx1250) — compile-verified
//
#include <hip/hip_runtime.h>
#include <hip/hip_bf16.h>
#include <math.h>

typedef __attribute__((ext_vector_type(2))) float v2f;
typedef __attribute__((ext_vector_type(8))) float v8f;

#define MODE_DIVCNT   0   // out = (acc + cnt*bias)/max(cnt,1)
#define MODE_RELU     1   // out = relu(acc + bias)
#define MODE_LOSS_POS 2   // lsum += softplus(relu(acc+bias)) - relu(...)
#define MODE_LOSS_NEG 3   // lsum += softplus(relu(acc+bias))

// ---------------------------------------------------------------------------
// Scatter pass 0: per edge e accumulate (by dst):
//   cnt[d] += 1, A[d] += nfeats[src], Bp[d] += ef[e], Bn[d] += ef[perm[e]]
// 16 threads per edge, float4 per thread (64 feats). Accumulators (25.6MB)
// live in the 192MB L2, so global_atomic_add_f32 stays on-chip.
// ---------------------------------------------------------------------------
__global__ void dgi_scatter0(const float* __restrict__ nfeats,
                             const float* __restrict__ ef,
                             const int* __restrict__ src,
                             const int* __restrict__ dst,
                             const int* __restrict__ perm,
                             float* __restrict__ cnt,
                             float* __restrict__ A,
                             float* __restrict__ Bp,
                             float* __restrict__ Bn,
                             int E)
{
    int idx = blockIdx.x * blockDim.x + threadIdx.x;
    int e = idx >> 4;
    if (e >= E) return;
    int c = (idx & 15) << 2;
    int s = src[e], d = dst[e], p = perm[e];
    float4 hv = *(const float4*)(nfeats + (size_t)s * 64 + c);
    float4 ev = *(const float4*)(ef + (size_t)e * 64 + c);
    float4 en = *(const float4*)(ef + (size_t)p * 64 + c);
    float* a  = A  + (size_t)d * 64 + c;
    float* bp = Bp + (size_t)d * 64 + c;
    float* bn = Bn + (size_t)d * 64 + c;
    atomicAdd(a + 0, hv.x); atomicAdd(a + 1, hv.y);
    atomicAdd(a + 2, hv.z); atomicAdd(a + 3, hv.w);
    atomicAdd(bp + 0, ev.x); atomicAdd(bp + 1, ev.y);
    atomicAdd(bp + 2, ev.z); atomicAdd(bp + 3, ev.w);
    atomicAdd(bn + 0, en.x); atomicAdd(bn + 1, en.y);
    atomicAdd(bn + 2, en.z); atomicAdd(bn + 3, en.w);
    if (c == 0) atomicAdd(cnt + d, 1.0f);
}

// Scatter pass 1: A_P[d] += h0p[src], A_N[d] += h0n[src]
__global__ void dgi_scatter1(const float* __restrict__ hp,
                             const float* __restrict__ hn,
                             const int* __restrict__ src,
                             const int* __restrict__ dst,
                             float* __restrict__ AP,
                             float* __restrict__ AN,
                             int E)
{
    int idx = blockIdx.x * blockDim.x + threadIdx.x;
    int e = idx >> 4;
    if (e >= E) return;
    int c = (idx & 15) << 2;
    int s = src[e], d = dst[e];
    float4 vp = *(const float4*)(hp + (size_t)s * 64 + c);
    float4 vn = *(const float4*)(hn + (size_t)s * 64 + c);
    float* ap = AP + (size_t)d * 64 + c;
    float* an = AN + (size_t)d * 64 + c;
    atomicAdd(ap + 0, vp.x); atomicAdd(ap + 1, vp.y);
    atomicAdd(ap + 2, vp.z); atomicAdd(ap + 3, vp.w);
    atomicAdd(an + 0, vn.x); atomicAdd(an + 1, vn.y);
    atomicAdd(an + 2, vn.z); atomicAdd(an + 3, vn.w);
}

// ---------------------------------------------------------------------------
// Fused dual-input GEMM, one wave per 16-row x 64-col tile:
//   acc = X1 @ W[:, 0:64]^T + X2 @ W[:, 64:128]^T      (W is [64 x 128] row-major)
// then a mode-dependent epilogue. K=128 -> 32 v_wmma_f32_16x16x4_f32 steps
// per N-tile, 4 N-tiles => 128 WMMAs per wave.
// Requires Nnodes % 16 == 0 (true here: 100000).
// ---------------------------------------------------------------------------
__global__ void dgi_gemm_dual(const float* __restrict__ X1,
                              const float* __restrict__ X2,
                              const float* __restrict__ W,
                              const float* __restrict__ bias,
                              const float* __restrict__ cnt,
                              float* __restrict__ out,
                              float* __restrict__ loss_out,
                              int mode, float lscale, int Nnodes)
{
    int lane  = threadIdx.x & 31;
    int wave  = threadIdx.x >> 5;
    int tile  = blockIdx.x * (blockDim.x >> 5) + wave;
    if (tile * 16 >= Nnodes) return;          // wave-uniform: EXEC all-ones for WMMA
    int row0  = tile * 16;
    int l16   = lane & 15;
    int khalf = (lane >> 4) << 1;             // K sub-offset 0 or 2 (f32 16x16x4 A/B layout)

    v8f acc[4] = {};                          // 4 x (16x16 f32 C/D frags) = 64 output cols

    const float* Xh[2] = {X1, X2};
    #pragma unroll
    for (int h = 0; h < 2; ++h) {
        const float* xrow = Xh[h] + (size_t)(row0 + l16) * 64 + khalf;
        #pragma unroll
        for (int k = 0; k < 64; k += 4) {
            v2f a;
            a.x = xrow[k];
            a.y = xrow[k + 1];
            #pragma unroll
            for (int nt = 0; nt < 4; ++nt) {
                // B[k][n] = W[n][h*64 + k]  (transposed weight access)
                const float* wrow = W + (size_t)(nt * 16 + l16) * 128 + h * 64 + k + khalf;
                v2f b;
                b.x = wrow[0];
                b.y = wrow[1];
                acc[nt] = __builtin_amdgcn_wmma_f32_16x16x4_f32(
                    false, a, false, b, (short)0, acc[nt], false, false);
            }
        }
    }

    // Epilogue. C/D frag layout: VGPR v holds (M = v + 8*(lane>=16), N = lane&15).
    int rhalf = (lane >> 4) << 3;
    float lsum = 0.0f;
    #pragma unroll
    for (int v = 0; v < 8; ++v) {
        int r = row0 + rhalf + v;
        float cc = (mode == MODE_DIVCNT) ? cnt[r] : 0.0f;
        float rcp = (mode == MODE_DIVCNT) ? (1.0f / fmaxf(cc, 1.0f)) : 0.0f;
        #pragma unroll
        for (int nt = 0; nt < 4; ++nt) {
            int col = nt * 16 + l16;
            float bcol = bias[col];
            float val = acc[nt][v];
            if (mode == MODE_DIVCNT) {
                val = (val + cc * bcol) * rcp;      // mean-aggregate (+ per-edge bias)
                out[(size_t)r * 64 + col] = val;
            } else {
                val = fmaxf(val + bcol, 0.0f);      // relu
                if (mode == MODE_RELU) {
                    out[(size_t)r * 64 + col] = val;
                } else {
                    // stable softplus = logaddexp(0, val); val >= 0 here
                    float sp = val + log1pf(expf(-val));
                    lsum += (mode == MODE_LOSS_POS) ? (sp - val) : sp;
                }
            }
        }
    }

    if (mode >= MODE_LOSS_POS) {
        #pragma unroll
        for (int off = 16; off > 0; off >>= 1)
            lsum += __shfl_xor(lsum, off, 32);
        if (lane == 0) atomicAdd(loss_out, lsum * lscale);
    }
}

extern "C" void kernel_launch(void* const* d_in, const int* in_sizes, int n_in,
                              void* d_out, int out_size, void* d_ws, size_t ws_size,
                              hipStream_t stream)
{
    const float* nfeats = (const float*)d_in[0];
    const float* efeats = (const float*)d_in[1];
    const int*   src    = (const int*)d_in[2];
    const int*   dst    = (const int*)d_in[3];
    const int*   perm   = (const int*)d_in[4];
    const float* Wm0 = (const float*)d_in[5];
    const float* bm0 = (const float*)d_in[6];
    const float* Wa0 = (const float*)d_in[7];
    const float* ba0 = (const float*)d_in[8];
    const float* Wm1 = (const float*)d_in[9];
    const float* bm1 = (const float*)d_in[10];
    const float* Wa1 = (const float*)d_in[11];
    const float* ba1 = (const float*)d_in[12];

    const int N = in_sizes[0] / 64;
    const int E = in_sizes[2];
    const size_t NB = (size_t)N * 64;

    float* ws  = (float*)d_ws;
    float* cnt = ws;            // N
    float* A_P = ws + N;        // NB   (layer-0 neighbor sum of nfeats; reused as layer-1 pos)
    float* A_N = A_P + NB;      // NB   (layer-1 neg neighbor sum)
    float* Bp  = A_N + NB;      // NB   (edge-feat sums, positive)
    float* Bn  = Bp + NB;       // NB   (edge-feat sums, permuted/negative)
    float* H0P = Bn + NB;       // NB
    float* H0N = H0P + NB;      // NB
    float* HN  = H0N + NB;      // NB   (scratch h_neigh, reused for all 4 combines)
    float* lossv = (float*)d_out;

    // zero cnt, A_P, A_N, Bp, Bn and the output scalar (capture-safe)
    hipMemsetAsync(ws, 0, ((size_t)N + 4 * NB) * sizeof(float), stream);
    hipMemsetAsync(d_out, 0, sizeof(float), stream);

    const int sblocks = (E * 16 + 255) / 256;
    dgi_scatter0<<<sblocks, 256, 0, stream>>>(nfeats, efeats, src, dst, perm,
                                              cnt, A_P, Bp, Bn, E);

    const int tiles   = (N + 15) / 16;
    const int gblocks = (tiles + 7) / 8;   // 8 waves (tiles) per 256-thread block

    // ---- layer 0, positive ----
    dgi_gemm_dual<<<gblocks, 256, 0, stream>>>(A_P, Bp, Wm0, bm0, cnt, HN,  nullptr, MODE_DIVCNT, 0.f, N);
    dgi_gemm_dual<<<gblocks, 256, 0, stream>>>(nfeats, HN, Wa0, ba0, nullptr, H0P, nullptr, MODE_RELU, 0.f, N);
    // ---- layer 0, negative ----
    dgi_gemm_dual<<<gblocks, 256, 0, stream>>>(A_P, Bn, Wm0, bm0, cnt, HN,  nullptr, MODE_DIVCNT, 0.f, N);
    dgi_gemm_dual<<<gblocks, 256, 0, stream>>>(nfeats, HN, Wa0, ba0, nullptr, H0N, nullptr, MODE_RELU, 0.f, N);

    // ---- layer 1 neighbor sums ----
    hipMemsetAsync(A_P, 0, 2 * NB * sizeof(float), stream);   // A_P, A_N contiguous
    dgi_scatter1<<<sblocks, 256, 0, stream>>>(H0P, H0N, src, dst, A_P, A_N, E);

    const float lscale = 1.0f / ((float)N * 64.0f);
    // ---- layer 1, positive (apply fused with BCE target=1) ----
    dgi_gemm_dual<<<gblocks, 256, 0, stream>>>(A_P, Bp, Wm1, bm1, cnt, HN, nullptr, MODE_DIVCNT, 0.f, N);
    dgi_gemm_dual<<<gblocks, 256, 0, stream>>>(H0P, HN, Wa1, ba1, nullptr, nullptr, lossv, MODE_LOSS_POS, lscale, N);
    // ---- layer 1, negative (apply fused with BCE target=0) ----
    dgi_gemm_dual<<<gblocks, 256, 0, stream>>>(A_N, Bn, Wm1, bm1, cnt, HN, nullptr, MODE_DIVCNT, 0.f, N);
    dgi_gemm_dual<<<gblocks, 256, 0, stream>>>(H0N, HN, Wa1, ba1, nullptr, nullptr, lossv, MODE_LOSS_NEG, lscale, N);
}